// GraphIoT_87179246174626
// MI455X (gfx1250) — compile-verified
//
#include <hip/hip_runtime.h>
#include <hip/hip_bf16.h>

typedef __attribute__((ext_vector_type(2))) float v2f;
typedef __attribute__((ext_vector_type(8))) float v8f;

#define RANK  64
#define NORD  3
#define MFL   50
#define NCLS  10
#define BN_EPS 1e-5f

// ---------------------------------------------------------------- zero ro
__global__ __launch_bounds__(256) void zero_kernel(float4* p, int n4) {
    int t = blockIdx.x * 256 + threadIdx.x;
    if (t < n4) p[t] = make_float4(0.f, 0.f, 0.f, 0.f);
}

// ---------------- h[n,r] = feats[n]*W_seq[r] + b_seq[r];  xbuf = h (dual write)
__global__ __launch_bounds__(256) void init_h_kernel(
    const float* __restrict__ feats, const float* __restrict__ W_seq,
    const float* __restrict__ b_seq, float* __restrict__ h,
    float* __restrict__ xbuf) {
    int t  = blockIdx.x * 256 + threadIdx.x;   // N*16 threads, float4 each
    int n  = t >> 4;
    int c4 = t & 15;
    float f = feats[n];
    float4 w = ((const float4*)W_seq)[c4];
    float4 b = ((const float4*)b_seq)[c4];
    float4 o;
    o.x = fmaf(f, w.x, b.x); o.y = fmaf(f, w.y, b.y);
    o.z = fmaf(f, w.z, b.z); o.w = fmaf(f, w.w, b.w);
    ((float4*)(h    + (size_t)n * RANK))[c4] = o;
    ((float4*)(xbuf + (size_t)n * RANK))[c4] = o;
}

// ---------------------------------------- xbuf[dst] += h[src]  (GIN sum aggregation)
__global__ __launch_bounds__(256) void scatter_kernel(
    const float* __restrict__ h, float* __restrict__ xbuf,
    const int* __restrict__ src, const int* __restrict__ dst) {
    int t = blockIdx.x * 256 + threadIdx.x;    // E*16 threads
    int e = t >> 4;
    int c = (t & 15) * 4;
    int s = src[e], d = dst[e];
    float4 v = *(const float4*)(h + (size_t)s * RANK + c);
    float* p = xbuf + (size_t)d * RANK + c;
    unsafeAtomicAdd(p + 0, v.x);   // global_atomic_add_f32, L2-resident target
    unsafeAtomicAdd(p + 1, v.y);
    unsafeAtomicAdd(p + 2, v.z);
    unsafeAtomicAdd(p + 3, v.w);
}

// -------------------- fused 3-layer MLP (WMMA f32) + BN fold + readout accumulate
// block = 256 threads = 8 waves; block tile = 32 nodes x 64 feats
// wave (mt,nt): 16x16 output tile, 16 x V_WMMA_F32_16X16X4_F32 per layer
#define XSTR 66   // LDS row stride (floats), padded: bank step 2/row
__global__ __launch_bounds__(256) void mlp_kernel(
    const float* __restrict__ xbuf, float* __restrict__ hout,
    float* __restrict__ xnext,     // pre-seed next order's scatter target with h
    float* __restrict__ ro,
    const float* __restrict__ W_lin, const float* __restrict__ b_lin,
    const float* __restrict__ bn_g, const float* __restrict__ bn_b,
    const float* __restrict__ bn_m, const float* __restrict__ bn_v,
    int order) {
    __shared__ float sW[3 * RANK * RANK];     // 48 KB: all 3 layer weights
    __shared__ float sX[2][32 * XSTR];        // ping-pong activation tiles

    const int tid = threadIdx.x;
    const int node0 = blockIdx.x * 32;

    // stage weights for this order (12288 floats) and the input tile
    {
        const float4* wsrc = (const float4*)(W_lin + (size_t)order * 3 * RANK * RANK);
        float4* wdst = (float4*)sW;
        #pragma unroll
        for (int i = tid; i < 3072; i += 256) wdst[i] = wsrc[i];
        for (int i = tid; i < 512; i += 256) {           // 32 rows x 16 float4
            int row = i >> 4, c4 = i & 15;
            float4 v = ((const float4*)(xbuf + (size_t)(node0 + row) * RANK))[c4];
            float* d = &sX[0][row * XSTR + c4 * 4];
            d[0] = v.x; d[1] = v.y; d[2] = v.z; d[3] = v.w;
        }
    }
    __syncthreads();

    const int wave = tid >> 5, lane = tid & 31;
    const int half = lane >> 4, lm = lane & 15;
    const int mt = wave >> 2, nt = wave & 3;
    const int n0 = nt * 16;
    const int arow = (mt * 16 + lm) * XSTR;

    int cur = 0;
    #pragma unroll
    for (int j = 0; j < 3; ++j) {
        v8f acc = {};
        const float* Wj = &sW[j * RANK * RANK];
        const float* Xi = &sX[cur][0];
        #pragma unroll
        for (int k = 0; k < RANK; k += 4) {
            const int kk = k + 2 * half;
            v2f a, b;
            a.x = Xi[arow + kk];                // A 16x4: lanes 0-15 K=0/1, 16-31 K=2/3
            a.y = Xi[arow + kk + 1];
            b.x = Wj[kk * RANK + n0 + lm];      // B 4x16: row K striped over lanes
            b.y = Wj[(kk + 1) * RANK + n0 + lm];
            acc = __builtin_amdgcn_wmma_f32_16x16x4_f32(
                false, a, false, b, (short)0, acc, false, false);
        }
        // epilogue: +bias, ReLU, folded eval-BN  (out = relu(z)*s + t)
        const int col  = n0 + lm;
        const int pofs = (order * 3 + j) * RANK + col;
        const float bb = b_lin[pofs];
        const float s  = bn_g[pofs] * rsqrtf(bn_v[pofs] + BN_EPS);
        const float tt = fmaf(-bn_m[pofs], s, bn_b[pofs]);
        if (j < 2) {
            float* Xo = &sX[cur ^ 1][0];
            #pragma unroll
            for (int r = 0; r < 8; ++r) {        // C/D: row = r + 8*half, col = lane%16
                float v = fmaxf(acc[r] + bb, 0.f);
                Xo[(mt * 16 + half * 8 + r) * XSTR + col] = fmaf(v, s, tt);
            }
            __syncthreads();
            cur ^= 1;
        } else {
            #pragma unroll
            for (int r = 0; r < 8; ++r) {
                float v = fmaxf(acc[r] + bb, 0.f);
                v = fmaf(v, s, tt);
                int node = node0 + mt * 16 + half * 8 + r;
                hout [(size_t)node * RANK + col] = v;
                xnext[(size_t)node * RANK + col] = v;   // seed next scatter target
                unsafeAtomicAdd(&ro[(node / MFL) * (NORD * RANK) + order * RANK + col], v);
            }
        }
    }
}

// ------------------------------------- y[g,c] = ro[g,:] @ W_cls + b_cls
__global__ __launch_bounds__(256) void cls_kernel(
    const float* __restrict__ ro, const float* __restrict__ W_cls,
    const float* __restrict__ b_cls, float* __restrict__ y, int total) {
    int t = blockIdx.x * 256 + threadIdx.x;
    if (t >= total) return;
    int g = t / NCLS, c = t % NCLS;
    float acc = b_cls[c];
    const float* r = ro + (size_t)g * (NORD * RANK);
    #pragma unroll 4
    for (int k = 0; k < NORD * RANK; ++k) acc = fmaf(r[k], W_cls[k * NCLS + c], acc);
    y[t] = acc;
}

extern "C" void kernel_launch(void* const* d_in, const int* in_sizes, int n_in,
                              void* d_out, int out_size, void* d_ws, size_t ws_size,
                              hipStream_t stream) {
    const float* feats = (const float*)d_in[0];
    const int*   src   = (const int*)d_in[1];
    const int*   dst   = (const int*)d_in[2];
    const float* W_seq = (const float*)d_in[3];
    const float* b_seq = (const float*)d_in[4];
    const float* W_lin = (const float*)d_in[5];
    const float* b_lin = (const float*)d_in[6];
    const float* bn_g  = (const float*)d_in[7];
    const float* bn_b  = (const float*)d_in[8];
    const float* bn_m  = (const float*)d_in[9];
    const float* bn_v  = (const float*)d_in[10];
    const float* W_cls = (const float*)d_in[11];
    const float* b_cls = (const float*)d_in[12];
    float* y = (float*)d_out;

    const int N  = in_sizes[0];          // 204800
    const int E  = in_sizes[1];          // 3276800
    const int bs = N / MFL;              // 4096

    float* h  = (float*)d_ws;            // [N,64]
    float* xb = h  + (size_t)N * RANK;   // [N,64]
    float* ro = xb + (size_t)N * RANK;   // [bs, 192]

    const int ro4 = bs * NORD * RANK / 4;
    zero_kernel<<<(ro4 + 255) / 256, 256, 0, stream>>>((float4*)ro, ro4);
    init_h_kernel<<<N / 16, 256, 0, stream>>>(feats, W_seq, b_seq, h, xb);

    for (int i = 0; i < NORD; ++i) {
        scatter_kernel<<<E / 16, 256, 0, stream>>>(h, xb, src, dst);
        mlp_kernel<<<N / 32, 256, 0, stream>>>(xb, h, xb, ro, W_lin, b_lin,
                                               bn_g, bn_b, bn_m, bn_v, i);
    }
    cls_kernel<<<(bs * NCLS + 255) / 256, 256, 0, stream>>>(ro, W_cls, b_cls, y, bs * NCLS);
}